// PoolingByMultiheadAttention_25520695673454
// MI455X (gfx1250) — compile-verified
//
#include <hip/hip_runtime.h>
#include <math.h>

constexpr int   cE  = 1024;   // embed
constexpr int   cH  = 16;     // heads
constexpr int   cHD = 64;     // head dim
constexpr int   cB  = 32;     // batch
constexpr int   cS  = 2048;   // seq
constexpr int   cNC = 8;      // chunks per batch (split-S)
constexpr int   cCS = cS / cNC;   // 256 tokens per chunk
constexpr float cLN_EPS = 1e-5f;
constexpr float cSCALE  = 0.125f; // 1/sqrt(64)

typedef __attribute__((ext_vector_type(2))) float v2f;
typedef __attribute__((ext_vector_type(8))) float v8f;

// ---------------- Kernel A1: q = seed @ wq.T + bq -------------------------
__global__ void k_qproj(const float* __restrict__ seed,
                        const float* __restrict__ w,      // in_proj_w [3E][E]
                        const float* __restrict__ bias,   // in_proj_b [3E]
                        float* __restrict__ q) {
  int i = blockIdx.x * blockDim.x + threadIdx.x;          // 0..1023
  const float4* row = (const float4*)(w + (size_t)i * cE);
  const float4* sd  = (const float4*)seed;
  float acc = 0.f;
  for (int e = 0; e < cE / 4; ++e) {
    float4 sv = sd[e];
    float4 wv = row[e];
    acc += sv.x * wv.x + sv.y * wv.y + sv.z * wv.z + sv.w * wv.w;
  }
  q[i] = acc + bias[i];
}

// ---------------- Kernel A2: fold q into Wk, pack as WMMA B operand -------
// U[e][h] = sum_d q[h*64+d]*wk[h*64+d][e], stored so that each lane's
// float2 B-fragment for K-group g is one contiguous b64 load:
//   lane = (e&2 ? 16 : 0) + h, slot = e&1, Upack[((e>>2)*32 + lane)*2 + slot]
__global__ void k_fold_k(const float* __restrict__ q,
                         const float* __restrict__ w,
                         const float* __restrict__ bias,
                         float* __restrict__ Upack,        // [E/4][32][2]
                         float* __restrict__ cvec) {       // [H]
  int t = blockIdx.x * blockDim.x + threadIdx.x;           // 0..16383
  int h = t >> 10;
  int e = t & (cE - 1);
  const float* qh = q + h * cHD;
  float acc = 0.f;
  for (int d = 0; d < cHD; ++d)
    acc += qh[d] * w[(size_t)(cE + h * cHD + d) * cE + e];
  int lane = ((e & 2) ? 16 : 0) + h;
  Upack[(((size_t)(e >> 2) * 32 + lane) << 1) + (e & 1)] = acc;
  if (blockIdx.x == 0 && threadIdx.x < cH) {
    const float* qq = q + threadIdx.x * cHD;
    const float* bk = bias + cE + threadIdx.x * cHD;
    float c = 0.f;
    for (int d = 0; d < cHD; ++d) c += qq[d] * bk[d];
    cvec[threadIdx.x] = c;
  }
}

// ---------------- Fused flash-style chunk kernel --------------------------
// One block per (b, chunk of 256 tokens). x chunk (1 MB) is read for the
// score pass and re-read L2-hot for the weighted-sum pass: x crosses HBM
// only once overall (~256 MB instead of 512 MB).
__global__ void __launch_bounds__(256)
k_attn_partial(const float* __restrict__ x,               // [B][S][E]
               const int*   __restrict__ mask,            // [B][S]
               const float* __restrict__ Upack,           // packed B operand
               const float* __restrict__ cvec,            // [H]
               float* __restrict__ m_g,                   // [B*NC][H]
               float* __restrict__ l_g,                   // [B*NC][H]
               float* __restrict__ pacc) {                // [B*NC][H][E]
  __shared__ float Ulds[(cE / 4) * 32 * 2];               // 64 KB
  __shared__ float sc[cCS * cH];                          // 16 KB  [tok][h]
  __shared__ float pk[(cCS / 4) * 32 * 2];                // 16 KB packed A op
  __shared__ float red[256];
  __shared__ float msafe[cH];

  int chunk = blockIdx.x;                                 // 0..B*NC-1
  int b = chunk >> 3;                                     // NC = 8
  int c = chunk & (cNC - 1);
  int sBase = c * cCS;
  int wave = threadIdx.x >> 5;
  int lane = threadIdx.x & 31;
  int nl = lane & 15;
  int kh = (lane >> 4) << 1;                              // 0 or 2
  int mOff = (lane >> 4) * 8;

  // stage packed U into LDS via CDNA5 async DMA loads (no VGPR round-trip);
  // tracked by ASYNCcnt, completed with s_wait_asynccnt before the barrier.
  {
    const float4* src = (const float4*)Upack;
    unsigned int ldsBase = (unsigned int)(uintptr_t)(&Ulds[0]);
    for (int i = threadIdx.x; i < (cE / 4) * 32 * 2 / 4; i += 256) {
      const float4* gp = src + i;
      unsigned int loff = ldsBase + (unsigned int)(i * 16);
      asm volatile("global_load_async_to_lds_b128 %0, %1, off"
                   :: "v"(loff), "v"(gp) : "memory");
    }
    asm volatile("s_wait_asynccnt 0x0" ::: "memory");
  }
  __syncthreads();

  // ---- Phase 1: scores for 2 token-tiles per wave (16 tok x 16 heads) ----
  for (int t2 = 0; t2 < 2; ++t2) {
    int tloc = (wave * 2 + t2) * 16;
    const float* xrow = x + ((size_t)b * cS + sBase + tloc + nl) * cE + kh;
    v8f acc = {};
    for (int k = 0; k < cE; k += 4) {
      float2 av = *(const float2*)(xrow + k);
      v2f a; a.x = av.x; a.y = av.y;
      v2f bm = *(const v2f*)(Ulds + (((k >> 2) * 32 + lane) << 1));
      acc = __builtin_amdgcn_wmma_f32_16x16x4_f32(
          false, a, false, bm, (short)0, acc, false, false);
    }
    float ch = cvec[nl];
    for (int r = 0; r < 8; ++r) {
      int tl = tloc + mOff + r;
      float v = acc[r] * cSCALE + ch;
      if (mask[(size_t)b * cS + sBase + tl] == 0) v = -INFINITY;
      sc[tl * cH + nl] = v;
    }
  }
  __syncthreads();

  // ---- Phase 2: per-head chunk max / exp / sum; pack p as WMMA A operand -
  int part = threadIdx.x >> 4;                            // 16 parts x 16 tok
  int h = threadIdx.x & 15;
  {
    float lm = -INFINITY;
    for (int t = 0; t < 16; ++t)
      lm = fmaxf(lm, sc[(part * 16 + t) * cH + h]);
    red[part * 16 + h] = lm;
  }
  __syncthreads();
  if (threadIdx.x < cH) {
    float m = -INFINITY;
    for (int p2 = 0; p2 < 16; ++p2)
      m = fmaxf(m, red[p2 * 16 + threadIdx.x]);
    m_g[(size_t)chunk * cH + threadIdx.x] = m;
    msafe[threadIdx.x] = (m == -INFINITY) ? 0.f : m;      // fully-masked chunk
  }
  __syncthreads();
  {
    float ms = msafe[h];
    float ls = 0.f;
    for (int t = 0; t < 16; ++t) {
      int tok = part * 16 + t;
      float p = __expf(sc[tok * cH + h] - ms);
      ls += p;
      int r = tok & 3;
      int pl = ((r & 2) ? 16 : 0) + h;
      pk[(((tok >> 2) * 32 + pl) << 1) + (r & 1)] = p;
    }
    red[part * 16 + h] = ls;
  }
  __syncthreads();
  if (threadIdx.x < cH) {
    float l = 0.f;
    for (int p2 = 0; p2 < 16; ++p2) l += red[p2 * 16 + threadIdx.x];
    l_g[(size_t)chunk * cH + threadIdx.x] = l;
  }
  __syncthreads();

  // ---- Phase 3: partial ctx = p^T x over chunk (x re-read L2-hot) --------
  for (int et = 0; et < 8; ++et) {
    int ebase = (wave * 8 + et) * 16;
    const float* xcol = x + ((size_t)b * cS + sBase) * cE + ebase + nl;
    v8f acc = {};
    for (int k = 0; k < cCS; k += 4) {
      v2f a = *(const v2f*)(pk + (((k >> 2) * 32 + lane) << 1)); // ds_load_b64
      v2f bm;
      bm.x = xcol[(size_t)(k + kh) * cE];
      bm.y = xcol[(size_t)(k + kh + 1) * cE];
      acc = __builtin_amdgcn_wmma_f32_16x16x4_f32(
          false, a, false, bm, (short)0, acc, false, false);
    }
    for (int r = 0; r < 8; ++r) {
      int hh = mOff + r;
      pacc[((size_t)chunk * cH + hh) * cE + ebase + nl] = acc[r];
    }
  }
}

// ---------------- Combine scale: w_{b,c,h} = exp(m_c - M) / L -------------
__global__ void k_cscale(const float* __restrict__ m_g,
                         const float* __restrict__ l_g,
                         float* __restrict__ wsc) {
  int t = blockIdx.x * blockDim.x + threadIdx.x;          // 0..511
  if (t >= cB * cH) return;
  int b = t >> 4, h = t & 15;
  float M = -INFINITY;
  for (int c = 0; c < cNC; ++c)
    M = fmaxf(M, m_g[(size_t)(b * cNC + c) * cH + h]);
  float L = 0.f;
  for (int c = 0; c < cNC; ++c)
    L += __expf(m_g[(size_t)(b * cNC + c) * cH + h] - M) *
         l_g[(size_t)(b * cNC + c) * cH + h];
  float inv = 1.f / L;
  for (int c = 0; c < cNC; ++c)
    wsc[(size_t)(b * cNC + c) * cH + h] =
        __expf(m_g[(size_t)(b * cNC + c) * cH + h] - M) * inv;
}

// ---------------- Combine: y[b][h][e] = sum_c w * pacc_c ------------------
__global__ void __launch_bounds__(256)
k_combine(const float* __restrict__ pacc,
          const float* __restrict__ wsc,
          float* __restrict__ y) {
  int t = blockIdx.x * 256 + threadIdx.x;                 // B*H*E threads
  int e = t & (cE - 1);
  int bh = t >> 10;
  int b = bh >> 4, h = bh & 15;
  float acc = 0.f;
  for (int c = 0; c < cNC; ++c)
    acc += wsc[(size_t)(b * cNC + c) * cH + h] *
           pacc[((size_t)(b * cNC + c) * cH + h) * cE + e];
  y[t] = acc;
}

// ---------------- ctx[b][i] = Wv[i] . y[b][h(i)] + bv[i] ------------------
__global__ void k_vproj(const float* __restrict__ y,
                        const float* __restrict__ w,
                        const float* __restrict__ bias,
                        float* __restrict__ ctx) {
  int t = blockIdx.x * blockDim.x + threadIdx.x;          // 0..32767
  int b = t >> 10;
  int i = t & (cE - 1);
  int h = i >> 6;
  const float4* wrow = (const float4*)(w + (size_t)(2 * cE + i) * cE);
  const float4* yrow = (const float4*)(y + ((size_t)b * cH + h) * cE);
  float acc = 0.f;
  for (int e = 0; e < cE / 4; ++e) {
    float4 wv = wrow[e];
    float4 yv = yrow[e];
    acc += wv.x * yv.x + wv.y * yv.y + wv.z * yv.z + wv.w * yv.w;
  }
  ctx[(size_t)b * cE + i] = acc + bias[2 * cE + i];
}

// ---------------- out proj + LayerNorm, one block per batch row -----------
__global__ void __launch_bounds__(256)
k_out_ln(const float* __restrict__ ctx,
         const float* __restrict__ w,                     // out_w [E][E]
         const float* __restrict__ bias,
         const float* __restrict__ g,
         const float* __restrict__ be,
         float* __restrict__ out) {
  __shared__ float o_sm[cE];
  __shared__ float r1[8], r2[8];
  int b = blockIdx.x;
  const float4* crow = (const float4*)(ctx + (size_t)b * cE);
  float s1 = 0.f, s2 = 0.f;
  for (int j = threadIdx.x; j < cE; j += 256) {
    const float4* wrow = (const float4*)(w + (size_t)j * cE);
    float acc = 0.f;
    for (int i = 0; i < cE / 4; ++i) {
      float4 cv = crow[i];
      float4 wv = wrow[i];
      acc += cv.x * wv.x + cv.y * wv.y + cv.z * wv.z + cv.w * wv.w;
    }
    acc += bias[j];
    o_sm[j] = acc;
    s1 += acc;
    s2 += acc * acc;
  }
  for (int o = 16; o > 0; o >>= 1) {
    s1 += __shfl_down(s1, o, 32);
    s2 += __shfl_down(s2, o, 32);
  }
  if ((threadIdx.x & 31) == 0) { r1[threadIdx.x >> 5] = s1; r2[threadIdx.x >> 5] = s2; }
  __syncthreads();
  if (threadIdx.x < 32) {
    float a = (threadIdx.x < 8) ? r1[threadIdx.x] : 0.f;
    float c = (threadIdx.x < 8) ? r2[threadIdx.x] : 0.f;
    for (int o = 4; o > 0; o >>= 1) { a += __shfl_down(a, o, 32); c += __shfl_down(c, o, 32); }
    if (threadIdx.x == 0) { r1[0] = a; r2[0] = c; }
  }
  __syncthreads();
  float mu  = r1[0] / cE;
  float var = r2[0] / cE - mu * mu;
  float inv = rsqrtf(var + cLN_EPS);
  for (int j = threadIdx.x; j < cE; j += 256)
    out[(size_t)b * cE + j] = (o_sm[j] - mu) * inv * g[j] + be[j];
}

extern "C" void kernel_launch(void* const* d_in, const int* in_sizes, int n_in,
                              void* d_out, int out_size, void* d_ws, size_t ws_size,
                              hipStream_t stream) {
  (void)in_sizes; (void)n_in; (void)out_size; (void)ws_size;
  const float* x     = (const float*)d_in[0];
  const int*   mask  = (const int*)  d_in[1];
  const float* seed  = (const float*)d_in[2];
  const float* in_w  = (const float*)d_in[3];
  const float* in_b  = (const float*)d_in[4];
  const float* out_w = (const float*)d_in[5];
  const float* out_b = (const float*)d_in[6];
  const float* ln_g  = (const float*)d_in[7];
  const float* ln_b  = (const float*)d_in[8];

  char* ws = (char*)d_ws;
  float* q     = (float*)(ws);                            // 4 KB
  float* cvec  = (float*)(ws + 4096);                     // 64 B
  float* Upack = (float*)(ws + 8192);                     // 64 KB
  float* m_g   = (float*)(ws + (128 << 10));              // 16 KB
  float* l_g   = (float*)(ws + (128 << 10) + (16 << 10)); // 16 KB
  float* wsc   = (float*)(ws + (128 << 10) + (32 << 10)); // 16 KB
  float* pacc  = (float*)(ws + (256 << 10));              // 16 MB
  float* y     = (float*)(ws + (256 << 10) + ((size_t)cB * cNC * cH * cE * 4)); // 2 MB
  float* ctx   = (float*)(ws + (256 << 10) + ((size_t)cB * cNC * cH * cE * 4)
                             + ((size_t)cB * cH * cE * 4));                     // 128 KB

  k_qproj       <<<4,          256, 0, stream>>>(seed, in_w, in_b, q);
  k_fold_k      <<<64,         256, 0, stream>>>(q, in_w, in_b, Upack, cvec);
  k_attn_partial<<<cB * cNC,   256, 0, stream>>>(x, mask, Upack, cvec, m_g, l_g, pacc);
  k_cscale      <<<2,          256, 0, stream>>>(m_g, l_g, wsc);
  k_combine     <<<cB * cH * cE / 256, 256, 0, stream>>>(pacc, wsc, y);
  k_vproj       <<<cB * cE / 256,      256, 0, stream>>>(y, in_w, in_b, ctx);
  k_out_ln      <<<cB,         256, 0, stream>>>(ctx, out_w, out_b, ln_g, ln_b, (float*)d_out);
}